// MPNN_1_38792144617821
// MI455X (gfx1250) — compile-verified
//
#include <hip/hip_runtime.h>
#include <hip/hip_bf16.h>

// ---- problem constants (match reference) ----
#define N_NODES 50000
#define N_EDGES 800000
#define FEAT_DIM 64
#define NUM_RELS 8

typedef __attribute__((ext_vector_type(2))) float v2f;
typedef __attribute__((ext_vector_type(8))) float v8f;

// ---------------------------------------------------------------------------
// Kernel 0: zero the mailbox M [N,64] and deg [N] (contiguous in workspace)
// ---------------------------------------------------------------------------
__global__ void zero_ws_kernel(float* __restrict__ p, int n) {
    int gid = blockIdx.x * blockDim.x + threadIdx.x;
    int stride = gridDim.x * blockDim.x;
    for (int i = gid; i < n; i += stride) p[i] = 0.0f;
}

// ---------------------------------------------------------------------------
// Kernel 1: in-degree via float atomics (matches reference segment_sum of 1.0)
// ---------------------------------------------------------------------------
__global__ void deg_kernel(const int* __restrict__ dst, float* __restrict__ deg) {
    int e = blockIdx.x * blockDim.x + threadIdx.x;
    if (e < N_EDGES) atomicAdd(&deg[dst[e]], 1.0f);
}

// ---------------------------------------------------------------------------
// Kernel 2: hw[r,n,f] = sum_k F[n,k] * W[r,k,f]   via V_WMMA_F32_16X16X4_F32
// grid = (N_NODES/16, NUM_RELS), block = 128 (4 waves; wave w owns cols w*16..)
// A-frag (16x4 f32, ISA 7.12.2): lanes 0-15 -> M=lane, {K,K+1}; lanes 16-31 ->
// M=lane-16, {K+2,K+3}. B-frag mirrored (N across lanes). D: VGPR i holds
// row (i + 8*half), col = lane&15.
// ---------------------------------------------------------------------------
__global__ void gemm_hw_kernel(const float* __restrict__ F,
                               const float* __restrict__ W,
                               float* __restrict__ hw) {
    const int mtile = blockIdx.x;            // 0..3124
    const int r     = blockIdx.y;            // 0..7
    const int wave  = threadIdx.x >> 5;      // 0..3
    const int lane  = threadIdx.x & 31;
    const int half  = lane >> 4;             // 0: K+{0,1}, 1: K+{2,3}
    const int l15   = lane & 15;

    const int m0 = mtile * 16;
    const int n0 = wave * 16;
    const float* __restrict__ Wr = W + (size_t)r * FEAT_DIM * FEAT_DIM;

    const int arow = m0 + l15;               // A row for this lane
    const int bcol = n0 + l15;               // B col for this lane

    v8f acc = {};
#pragma unroll
    for (int k = 0; k < FEAT_DIM; k += 4) {
        const int ka = k + half * 2;
        v2f a, b;
        a.x = F[(size_t)arow * FEAT_DIM + ka + 0];
        a.y = F[(size_t)arow * FEAT_DIM + ka + 1];
        b.x = Wr[(size_t)(ka + 0) * FEAT_DIM + bcol];
        b.y = Wr[(size_t)(ka + 1) * FEAT_DIM + bcol];
        acc = __builtin_amdgcn_wmma_f32_16x16x4_f32(
            /*neg_a=*/false, a, /*neg_b=*/false, b,
            /*c_mod=*/(short)0, acc, /*reuse_a=*/false, /*reuse_b=*/false);
    }

#pragma unroll
    for (int i = 0; i < 8; ++i) {
        const int row = m0 + i + half * 8;
        hw[((size_t)r * N_NODES + row) * FEAT_DIM + n0 + l15] = acc[i];
    }
}

// ---------------------------------------------------------------------------
// Kernel 3: per edge-feature: M[dst,f] += hw[etype,src,f] / deg[dst]
// One thread per (edge, feature). hw (102.4 MB) is L2-resident on MI455X.
// ---------------------------------------------------------------------------
__global__ void edge_scatter_kernel(const int* __restrict__ src,
                                    const int* __restrict__ dst,
                                    const int* __restrict__ etype,
                                    const float* __restrict__ hw,
                                    const float* __restrict__ deg,
                                    float* __restrict__ M) {
    const int gid = blockIdx.x * blockDim.x + threadIdx.x;   // < E*64 = 51.2M
    const int e = gid >> 6;
    const int f = gid & 63;
    if (e >= N_EDGES) return;
    const int s = src[e];
    const int d = dst[e];
    const int r = etype[e];
    const float v = hw[((size_t)r * N_NODES + s) * FEAT_DIM + f] / deg[d];
    atomicAdd(&M[(size_t)d * FEAT_DIM + f], v);
}

// ---------------------------------------------------------------------------
// Kernel 4: out = deg>0 ? relu(0.2*h + 0.8*M) : h
// ---------------------------------------------------------------------------
__global__ void update_kernel(const float* __restrict__ F,
                              const float* __restrict__ M,
                              const float* __restrict__ deg,
                              float* __restrict__ out) {
    const int gid = blockIdx.x * blockDim.x + threadIdx.x;
    if (gid >= N_NODES * FEAT_DIM) return;
    const int n = gid >> 6;
    const float h = F[gid];
    float o = h;
    if (deg[n] > 0.0f) {
        const float u = 0.2f * h + 0.8f * M[gid];
        o = u > 0.0f ? u : 0.0f;
    }
    out[gid] = o;
}

// ---------------------------------------------------------------------------
extern "C" void kernel_launch(void* const* d_in, const int* in_sizes, int n_in,
                              void* d_out, int out_size, void* d_ws, size_t ws_size,
                              hipStream_t stream) {
    const float* F     = (const float*)d_in[0];   // [50000,64]
    const float* W     = (const float*)d_in[1];   // [8,64,64]
    const int*   src   = (const int*)d_in[2];     // [800000]
    const int*   dst   = (const int*)d_in[3];     // [800000]
    const int*   etype = (const int*)d_in[4];     // [800000]
    float* out = (float*)d_out;

    // workspace layout: hw | M | deg  (115.4 MB total)
    char* ws = (char*)d_ws;
    float* hw  = (float*)ws;                                            // 8*N*64
    float* M   = (float*)(ws + (size_t)NUM_RELS * N_NODES * FEAT_DIM * 4);
    float* deg = (float*)(ws + (size_t)NUM_RELS * N_NODES * FEAT_DIM * 4
                             + (size_t)N_NODES * FEAT_DIM * 4);

    // 0) zero M (+deg, contiguous): 50000*64 + 50000 floats
    const int nzero = N_NODES * FEAT_DIM + N_NODES;
    zero_ws_kernel<<<2048, 256, 0, stream>>>(M, nzero);

    // 1) in-degree
    deg_kernel<<<(N_EDGES + 255) / 256, 256, 0, stream>>>(dst, deg);

    // 2) relation GEMMs on the WMMA pipe (f32, full precision)
    dim3 ggrid(N_NODES / 16, NUM_RELS);
    gemm_hw_kernel<<<ggrid, 128, 0, stream>>>(F, W, hw);

    // 3) gather + normalize + scatter-add (L2-resident hw and M)
    const long long work = (long long)N_EDGES * FEAT_DIM;               // 51.2M
    edge_scatter_kernel<<<(int)((work + 255) / 256), 256, 0, stream>>>(
        src, dst, etype, hw, deg, M);

    // 4) final blend / relu / where(deg>0)
    update_kernel<<<(N_NODES * FEAT_DIM + 255) / 256, 256, 0, stream>>>(
        F, M, deg, out);
}